// GCN_46488726012137
// MI455X (gfx1250) — compile-verified
//
#include <hip/hip_runtime.h>
#include <hip/hip_bf16.h>

// ---------------- problem constants ----------------
#define NNODES 30000
#define NEDGES 480000
#define NFEATC 512
#define NHIDC  128
#define NCLS   16
#define EPSR   0.3f
#define NT     4        // 16x16 col tiles per wave (A reused NT times)

typedef __attribute__((ext_vector_type(2))) float v2f;
typedef __attribute__((ext_vector_type(8))) float v8f;

// fp32 WMMA: D(16x16) = A(16x4) * B(4x16) + C   (exact fp32 math)
__device__ __forceinline__ v8f wmma4(v2f a, v2f b, v8f c) {
  return __builtin_amdgcn_wmma_f32_16x16x4_f32(false, a, false, b, (short)0, c,
                                               false, false);
}

__device__ __forceinline__ float san(float v) { return (v != v) ? 0.0f : v; }

// ---------------- zero fill ----------------
__global__ void k_zero(float* __restrict__ p, size_t n) {
  size_t i = (size_t)blockIdx.x * blockDim.x + threadIdx.x;
  if (i < n) p[i] = 0.0f;
}

// ---------------- GEMM t1: mid1 = san(x) @ W^T + b ; raw1 = relu(mid1) ------
// grid: (N/16, 128/(16*NT)); one wave computes a 16x64 strip.
__global__ __launch_bounds__(32) void k_gemm_t1(
    const float* __restrict__ x, const float* __restrict__ W,
    const float* __restrict__ bias, float* __restrict__ mid1,
    float* __restrict__ raw1) {
  const int lane = threadIdx.x;
  const int half = lane >> 4;
  const int l16  = lane & 15;
  const int row  = blockIdx.x * 16 + l16;
  const int colb = blockIdx.y * (16 * NT);

  v8f acc[NT];
#pragma unroll
  for (int t = 0; t < NT; ++t)
#pragma unroll
    for (int g = 0; g < 8; ++g) acc[t][g] = 0.0f;

  const float* xr = x + (size_t)row * NFEATC;
  const float* wr[NT];
#pragma unroll
  for (int t = 0; t < NT; ++t)
    wr[t] = W + (size_t)(colb + 16 * t + l16) * NFEATC;   // W: [J,K] (JK)

#pragma unroll 4
  for (int k0 = 0; k0 < NFEATC; k0 += 4) {
    const int kb = k0 + half * 2;
    float2 xa = *(const float2*)(xr + kb);
    v2f a;
    a.x = san(xa.x); a.y = san(xa.y);
#pragma unroll
    for (int t = 0; t < NT; ++t) {
      float2 wb = *(const float2*)(wr[t] + kb);
      v2f b; b.x = wb.x; b.y = wb.y;
      acc[t] = wmma4(a, b, acc[t]);
    }
  }
#pragma unroll
  for (int t = 0; t < NT; ++t) {
    const int col = colb + 16 * t + l16;
    const float bv = bias[col];
#pragma unroll
    for (int g = 0; g < 8; ++g) {
      const int m = blockIdx.x * 16 + g + half * 8;
      const float v = acc[t][g] + bv;
      mid1[(size_t)m * NHIDC + col] = v;
      raw1[(size_t)m * NHIDC + col] = v > 0.0f ? v : 0.0f;
    }
  }
}

// ---- GEMM layer: out[:,0:128] = relu((EPS*raw + z) @ hw) , hw is [K,128] KJ
__global__ __launch_bounds__(32) void k_gemm_layer(
    const float* __restrict__ raw, const float* __restrict__ z, int K,
    const float* __restrict__ hw, float* __restrict__ outp, int ldc) {
  const int lane = threadIdx.x;
  const int half = lane >> 4;
  const int l16  = lane & 15;
  const int row  = blockIdx.x * 16 + l16;
  const int colb = blockIdx.y * (16 * NT);

  v8f acc[NT];
#pragma unroll
  for (int t = 0; t < NT; ++t)
#pragma unroll
    for (int g = 0; g < 8; ++g) acc[t][g] = 0.0f;

  const float* rr = raw + (size_t)row * K;
  const float* zr = z   + (size_t)row * K;
#pragma unroll 4
  for (int k0 = 0; k0 < K; k0 += 4) {
    const int kb = k0 + half * 2;
    float2 rv = *(const float2*)(rr + kb);
    float2 zv = *(const float2*)(zr + kb);
    v2f a;
    a.x = EPSR * rv.x + zv.x;
    a.y = EPSR * rv.y + zv.y;
    const float* h0 = hw + (size_t)kb * NHIDC + l16;        // KJ layout
    const float* h1 = hw + (size_t)(kb + 1) * NHIDC + l16;
#pragma unroll
    for (int t = 0; t < NT; ++t) {
      v2f b; b.x = h0[colb + 16 * t]; b.y = h1[colb + 16 * t];
      acc[t] = wmma4(a, b, acc[t]);
    }
  }
#pragma unroll
  for (int t = 0; t < NT; ++t) {
    const int col = colb + 16 * t + l16;
#pragma unroll
    for (int g = 0; g < 8; ++g) {
      const int m = blockIdx.x * 16 + g + half * 8;
      const float v = acc[t][g];
      outp[(size_t)m * ldc + col] = v > 0.0f ? v : 0.0f;
    }
  }
}

// ---- virtual concat A for t2: [h2(384) | raw1(128) | raw2(384) | san(x)(512)]
__device__ __forceinline__ float t2A(int n, int k, const float* __restrict__ h2,
                                     const float* __restrict__ raw1,
                                     const float* __restrict__ raw2,
                                     const float* __restrict__ x) {
  if (k < 384) return h2[(size_t)n * 384 + k];
  if (k < 512) return raw1[(size_t)n * 128 + (k - 384)];
  if (k < 896) return raw2[(size_t)n * 384 + (k - 512)];
  return san(x[(size_t)n * 512 + (k - 896)]);
}

// ---- GEMM t2: mid2 = concatA @ W_t2^T + b_t2   (no relu stored; K=1408) ----
__global__ __launch_bounds__(32) void k_gemm_t2(
    const float* __restrict__ h2, const float* __restrict__ raw1,
    const float* __restrict__ raw2, const float* __restrict__ x,
    const float* __restrict__ W, const float* __restrict__ bias,
    float* __restrict__ mid2) {
  const int lane = threadIdx.x;
  const int half = lane >> 4;
  const int l16  = lane & 15;
  const int row  = blockIdx.x * 16 + l16;
  const int colb = blockIdx.y * (16 * NT);
  const int K = 7 * NHIDC + NFEATC;   // 1408

  v8f acc[NT];
#pragma unroll
  for (int t = 0; t < NT; ++t)
#pragma unroll
    for (int g = 0; g < 8; ++g) acc[t][g] = 0.0f;

  const float* wr[NT];
#pragma unroll
  for (int t = 0; t < NT; ++t)
    wr[t] = W + (size_t)(colb + 16 * t + l16) * K;   // [J=128, K=1408] (JK)

  for (int k0 = 0; k0 < K; k0 += 4) {
    const int kb = k0 + half * 2;
    v2f a;
    a.x = t2A(row, kb, h2, raw1, raw2, x);
    a.y = t2A(row, kb + 1, h2, raw1, raw2, x);
#pragma unroll
    for (int t = 0; t < NT; ++t) {
      float2 wb = *(const float2*)(wr[t] + kb);
      v2f b; b.x = wb.x; b.y = wb.y;
      acc[t] = wmma4(a, b, acc[t]);
    }
  }
#pragma unroll
  for (int t = 0; t < NT; ++t) {
    const int col = colb + 16 * t + l16;
    const float bv = bias[col];
#pragma unroll
    for (int g = 0; g < 8; ++g) {
      const int m = blockIdx.x * 16 + g + half * 8;
      mid2[(size_t)m * NHIDC + col] = acc[t][g] + bv;   // pre-relu == mid2
    }
  }
}

// ---- GEMM t3: out = relu(mid2) @ W_t3^T + b_t3  (J=16, K=128) --------------
__global__ __launch_bounds__(32) void k_gemm_t3(
    const float* __restrict__ mid2, const float* __restrict__ W,
    const float* __restrict__ bias, float* __restrict__ outp) {
  const int lane = threadIdx.x;
  const int half = lane >> 4;
  const int l16  = lane & 15;
  const int row  = blockIdx.x * 16 + l16;
  const int col  = l16;   // 16 classes -> single col tile

  v8f acc;
#pragma unroll
  for (int g = 0; g < 8; ++g) acc[g] = 0.0f;

  const float* ar = mid2 + (size_t)row * NHIDC;
  const float* wr = W + (size_t)col * NHIDC;   // [J=16, K=128] (JK)
#pragma unroll 4
  for (int k0 = 0; k0 < NHIDC; k0 += 4) {
    const int kb = k0 + half * 2;
    v2f a, b;
    float2 av = *(const float2*)(ar + kb);
    a.x = av.x > 0.0f ? av.x : 0.0f;
    a.y = av.y > 0.0f ? av.y : 0.0f;
    float2 wb = *(const float2*)(wr + kb);
    b.x = wb.x; b.y = wb.y;
    acc = wmma4(a, b, acc);
  }
  const float bv = bias[col];
#pragma unroll
  for (int g = 0; g < 8; ++g) {
    const int m = blockIdx.x * 16 + g + half * 8;
    outp[(size_t)m * NCLS + col] = acc[g] + bv;
  }
}

// ---- edge-gated aggregation: z[dst] += d[dst]*d[src]*tanh(gate)*h[src] ----
template <int D>
__global__ __launch_bounds__(256) void k_edge(
    const float* __restrict__ h, const int* __restrict__ src,
    const int* __restrict__ dst, const float* __restrict__ dvec,
    const float* __restrict__ gw, const float* __restrict__ gb,
    float* __restrict__ z, int E) {
  constexpr int PERLANE = D / 32;      // 4 (D=128) or 12 (D=384)
  constexpr int NV4 = PERLANE / 4;     // 1 or 3 float4 per lane
  const int lane = threadIdx.x & 31;
  const int gwid = blockIdx.x * (blockDim.x >> 5) + (threadIdx.x >> 5);
  const int nw   = gridDim.x * (blockDim.x >> 5);
  const float gbv = gb[0];

  for (int e = gwid; e < E; e += nw) {
    const int s = src[e];
    const int t = dst[e];
    const float4* hs4  = (const float4*)(h + (size_t)s * D) + lane * NV4;
    const float4* ht4  = (const float4*)(h + (size_t)t * D) + lane * NV4;
    const float4* gwd4 = (const float4*)(gw)     + lane * NV4;  // gw[:D] w/ dst
    const float4* gws4 = (const float4*)(gw + D) + lane * NV4;  // gw[D:] w/ src
    float4 hs[NV4];
    float p = 0.0f;
#pragma unroll
    for (int i = 0; i < NV4; ++i) {
      hs[i] = hs4[i];
      const float4 ht = ht4[i];
      const float4 gd = gwd4[i];
      const float4 gs = gws4[i];
      p += ht.x * gd.x + ht.y * gd.y + ht.z * gd.z + ht.w * gd.w;
      p += hs[i].x * gs.x + hs[i].y * gs.y + hs[i].z * gs.z + hs[i].w * gs.w;
    }
#pragma unroll
    for (int off = 16; off > 0; off >>= 1) p += __shfl_xor(p, off, 32);
    const float a  = tanhf(p + gbv);
    const float ee = dvec[t] * dvec[s] * a;
    float* zr = z + (size_t)t * D + lane * PERLANE;
#pragma unroll
    for (int i = 0; i < NV4; ++i) {
      unsafeAtomicAdd(zr + 4 * i + 0, ee * hs[i].x);
      unsafeAtomicAdd(zr + 4 * i + 1, ee * hs[i].y);
      unsafeAtomicAdd(zr + 4 * i + 2, ee * hs[i].z);
      unsafeAtomicAdd(zr + 4 * i + 3, ee * hs[i].w);
    }
  }
}

// ---------------------------------------------------------------------------
extern "C" void kernel_launch(void* const* d_in, const int* in_sizes, int n_in,
                              void* d_out, int out_size, void* d_ws,
                              size_t ws_size, hipStream_t stream) {
  (void)in_sizes; (void)n_in; (void)out_size; (void)ws_size;

  const float* x = (const float*)d_in[0];
  const int*   srcs[3]; const int* dsts[3]; const float* dd[3];
  for (int i = 0; i < 3; ++i) {
    srcs[i] = (const int*)d_in[1 + 3 * i];
    dsts[i] = (const int*)d_in[2 + 3 * i];
    dd[i]   = (const float*)d_in[3 + 3 * i];
  }
  const float* W_t1 = (const float*)d_in[10];
  const float* b_t1 = (const float*)d_in[11];
  const float *gW1[3], *gb1[3], *hw1[3], *gW2[3], *gb2[3], *hw2[3];
  for (int i = 0; i < 3; ++i) {
    const int b = 12 + 6 * i;
    gW1[i] = (const float*)d_in[b + 0];
    gb1[i] = (const float*)d_in[b + 1];
    hw1[i] = (const float*)d_in[b + 2];
    gW2[i] = (const float*)d_in[b + 3];
    gb2[i] = (const float*)d_in[b + 4];
    hw2[i] = (const float*)d_in[b + 5];
  }
  const float* W_t2 = (const float*)d_in[30];
  const float* b_t2 = (const float*)d_in[31];
  const float* W_t3 = (const float*)d_in[32];
  const float* b_t3 = (const float*)d_in[33];

  // d_out = [out (N*16) | mid1 (N*128) | mid2 (N*128)]
  float* outp = (float*)d_out;
  float* mid1 = outp + (size_t)NNODES * NCLS;
  float* mid2 = mid1 + (size_t)NNODES * NHIDC;

  // workspace: raw1 | raw2 | h2 | z  (z reused for all six aggregations)
  float* raw1 = (float*)d_ws;
  float* raw2 = raw1 + (size_t)NNODES * NHIDC;
  float* h2b  = raw2 + (size_t)NNODES * 3 * NHIDC;
  float* zb   = h2b  + (size_t)NNODES * 3 * NHIDC;

  const dim3 gStrip(NNODES / 16, NHIDC / (16 * NT));   // 1875 x 2
  k_gemm_t1<<<gStrip, 32, 0, stream>>>(x, W_t1, b_t1, mid1, raw1);

  const int EB = 256;                       // 8 waves / block
  const int EG = NEDGES / (EB / 32);        // one wave per edge

  for (int i = 0; i < 3; ++i) {            // layer 1 (D = 128)
    const size_t zn = (size_t)NNODES * NHIDC;
    k_zero<<<(unsigned)((zn + 255) / 256), 256, 0, stream>>>(zb, zn);
    k_edge<128><<<EG, EB, 0, stream>>>(raw1, srcs[i], dsts[i], dd[i],
                                       gW1[i], gb1[i], zb, NEDGES);
    k_gemm_layer<<<gStrip, 32, 0, stream>>>(raw1, zb, NHIDC, hw1[i],
                                            raw2 + i * NHIDC, 3 * NHIDC);
  }
  for (int i = 0; i < 3; ++i) {            // layer 2 (D = 384)
    const size_t zn = (size_t)NNODES * 3 * NHIDC;
    k_zero<<<(unsigned)((zn + 255) / 256), 256, 0, stream>>>(zb, zn);
    k_edge<384><<<EG, EB, 0, stream>>>(raw2, srcs[i], dsts[i], dd[i],
                                       gW2[i], gb2[i], zb, NEDGES);
    k_gemm_layer<<<gStrip, 32, 0, stream>>>(raw2, zb, 3 * NHIDC, hw2[i],
                                            h2b + i * NHIDC, 3 * NHIDC);
  }

  k_gemm_t2<<<gStrip, 32, 0, stream>>>(h2b, raw1, raw2, x, W_t2, b_t2, mid2);

  const dim3 gT3(NNODES / 16, 1);
  k_gemm_t3<<<gT3, 32, 0, stream>>>(mid2, W_t3, b_t3, outp);
}